// ContrastiveLossWithRKD_65111704208064
// MI455X (gfx1250) — compile-verified
//
#include <hip/hip_runtime.h>
#include <hip/hip_bf16.h>

// ContrastiveLossWithRKD for MI455X (gfx1250, wave32).
// Gram-matrix reformulation: angle potentials e_ij.e_kj are computed from
// G = x x^T and precomputed 1/(||x_j-x_i||+eps) tables, turning a 52-GFLOP
// batched-GEMM tensor into a ~0.9-GFLOP coalesced elementwise reduction.
// The three remaining GEMMs run on V_WMMA_F32_16X16X4_F32 (fp32-exact).
// Angle kernel tiles 8 i-values per block so the i-independent coalesced
// streams are amortized (AI ~7 flop/B, VALU-bound).

#define EPSF 1e-8f
#define NR   256          // concat rows (2B)
#define DD   768          // feature dim
#define NB   128          // batch
#define IPB  8            // i-values per angle block
#define KPB  32           // k-values per angle block

typedef __attribute__((ext_vector_type(2))) float v2f;
typedef __attribute__((ext_vector_type(8))) float v8f;

__device__ __forceinline__ const float* sel_row(const float* __restrict__ q,
                                                const float* __restrict__ p, int r) {
    return (r < NB) ? (q + (size_t)r * DD) : (p + (size_t)(r - NB) * DD);
}

// -------- Gram: G[r][c] = <x_r, x_c>, x = concat(q,p) rows, 256x256 ---------
// 8 waves / block, one 16x16 output tile per wave, K=768 (192 WMMA f32 ops).
__global__ __launch_bounds__(256)
void gram_wmma_kernel(const float* __restrict__ q, const float* __restrict__ p,
                      float* __restrict__ G) {
    const int wave = threadIdx.x >> 5;
    const int lane = threadIdx.x & 31;
    const int tile = blockIdx.x * 8 + wave;          // 16x16 grid of tiles
    const int tm = (tile >> 4) << 4;
    const int tn = (tile & 15) << 4;
    const int hi = lane >> 4;
    const int lr = lane & 15;

    const float* arow = sel_row(q, p, tm + lr);
    const float* brow = sel_row(q, p, tn + lr);
    v8f acc = {0.f, 0.f, 0.f, 0.f, 0.f, 0.f, 0.f, 0.f};

#if __has_builtin(__builtin_amdgcn_wmma_f32_16x16x4_f32)
    const int ko = hi << 1;
    #pragma unroll 8
    for (int k = 0; k < DD; k += 4) {
        v2f a = *(const v2f*)(arow + k + ko);
        v2f b = *(const v2f*)(brow + k + ko);
        acc = __builtin_amdgcn_wmma_f32_16x16x4_f32(false, a, false, b,
                                                    (short)0, acc, false, false);
    }
#else
    const float* crow = sel_row(q, p, tn + lr);
    for (int r = 0; r < 8; ++r) {
        const float* rr = sel_row(q, p, tm + (hi << 3) + r);
        float s = 0.f;
        for (int k = 0; k < DD; ++k) s = fmaf(rr[k], crow[k], s);
        acc[r] = s;
    }
#endif

    const int col = tn + lr;
    const int rb  = tm + (hi << 3);
    #pragma unroll
    for (int r = 0; r < 8; ++r)
        G[(rb + r) * NR + col] = acc[r];
}

// -------- scores = sq @ sp^T  (128x128), same WMMA tiling -------------------
__global__ __launch_bounds__(256)
void scores_wmma_kernel(const float* __restrict__ sq, const float* __restrict__ sp,
                        float* __restrict__ S) {
    const int wave = threadIdx.x >> 5;
    const int lane = threadIdx.x & 31;
    const int tile = blockIdx.x * 8 + wave;          // 8x8 grid of tiles
    const int tm = (tile >> 3) << 4;
    const int tn = (tile & 7) << 4;
    const int hi = lane >> 4;
    const int lr = lane & 15;

    const float* arow = sq + (size_t)(tm + lr) * DD;
    const float* brow = sp + (size_t)(tn + lr) * DD;
    v8f acc = {0.f, 0.f, 0.f, 0.f, 0.f, 0.f, 0.f, 0.f};

#if __has_builtin(__builtin_amdgcn_wmma_f32_16x16x4_f32)
    const int ko = hi << 1;
    #pragma unroll 8
    for (int k = 0; k < DD; k += 4) {
        v2f a = *(const v2f*)(arow + k + ko);
        v2f b = *(const v2f*)(brow + k + ko);
        acc = __builtin_amdgcn_wmma_f32_16x16x4_f32(false, a, false, b,
                                                    (short)0, acc, false, false);
    }
#else
    const float* crow = sp + (size_t)(tn + lr) * DD;
    for (int r = 0; r < 8; ++r) {
        const float* rr = sq + (size_t)(tm + (hi << 3) + r) * DD;
        float s = 0.f;
        for (int k = 0; k < DD; ++k) s = fmaf(rr[k], crow[k], s);
        acc[r] = s;
    }
#endif

    const int col = tn + lr;
    const int rb  = tm + (hi << 3);
    #pragma unroll
    for (int r = 0; r < 8; ++r)
        S[(rb + r) * NB + col] = acc[r];
}

// -------- distance loss + inv-norm tables (single block, two phases) --------
__global__ __launch_bounds__(1024)
void distance_kernel(const float* __restrict__ Gs, const float* __restrict__ Gt,
                     float* __restrict__ invS, float* __restrict__ invT,
                     float* __restrict__ scal /* [0]=dist_loss */) {
    __shared__ float red[1024];
    __shared__ float means[2];
    const int t = threadIdx.x;

    float ss = 0.f, st = 0.f;
    for (int idx = t; idx < NR * NR; idx += 1024) {
        const int i = idx >> 8, j = idx & 255;
        float ds = fmaxf(Gs[i * NR + i] + Gs[j * NR + j] - 2.f * Gs[idx], 0.f);
        float dt = fmaxf(Gt[i * NR + i] + Gt[j * NR + j] - 2.f * Gt[idx], 0.f);
        invS[idx] = 1.f / (sqrtf(ds) + EPSF);
        invT[idx] = 1.f / (sqrtf(dt) + EPSF);
        if (i < j) { ss += ds; st += dt; }
    }
    red[t] = ss; __syncthreads();
    for (int o = 512; o > 0; o >>= 1) { if (t < o) red[t] += red[t + o]; __syncthreads(); }
    if (t == 0) means[0] = red[0] / 32640.f + EPSF;   // cnt = 256*255/2
    __syncthreads();
    red[t] = st; __syncthreads();
    for (int o = 512; o > 0; o >>= 1) { if (t < o) red[t] += red[t + o]; __syncthreads(); }
    if (t == 0) means[1] = red[0] / 32640.f + EPSF;
    __syncthreads();

    const float rms = 1.f / means[0], rmt = 1.f / means[1];
    float hl = 0.f;
    for (int idx = t; idx < NR * NR; idx += 1024) {
        const int i = idx >> 8, j = idx & 255;
        if (i < j) {
            float ds = fmaxf(Gs[i * NR + i] + Gs[j * NR + j] - 2.f * Gs[idx], 0.f);
            float dt = fmaxf(Gt[i * NR + i] + Gt[j * NR + j] - 2.f * Gt[idx], 0.f);
            float d = ds * rms - dt * rmt;
            float a = fabsf(d);
            hl += (a < 1.f) ? 0.5f * a * a : a - 0.5f;
        }
    }
    red[t] = hl; __syncthreads();
    for (int o = 512; o > 0; o >>= 1) { if (t < o) red[t] += red[t + o]; __syncthreads(); }
    if (t == 0) scal[0] = red[0] / 32640.f;
}

// -------- contrastive: mean_r( logsumexp(scores[r]/T) - scores[r][r]/T ) ----
__global__ __launch_bounds__(128)
void contrastive_kernel(const float* __restrict__ S, float* __restrict__ scal) {
    __shared__ float red[128];
    const int r = threadIdx.x;
    const float* row = S + r * NB;
    const float invT = 20.0f;                        // 1 / 0.05
    float m = -3.402823e38f;
    for (int c = 0; c < NB; ++c) m = fmaxf(m, row[c] * invT);
    float sum = 0.f;
    for (int c = 0; c < NB; ++c) sum += expf(row[c] * invT - m);
    red[r] = m + logf(sum) - row[r] * invT;
    __syncthreads();
    for (int o = 64; o > 0; o >>= 1) { if (r < o) red[r] += red[r + o]; __syncthreads(); }
    if (r == 0) scal[1] = red[0] / 128.f;
}

// -------- angle loss: sum over (i,j,k) of huber(ps - pt) --------------------
// ps[i,j,k] = (Gjj - Gij - Gjk + Gik) * inv[i][j] * inv[k][j]
// Tiling: block = (i-group of 8) x (k-chunk of 32), thread = j.
// Coalesced k-streams G[k][j], inv[k][j] are i-independent -> amortized 8x.
// G[i][k] tiles (j-independent) staged in LDS. i==j lanes cancel to exact 0.
__global__ __launch_bounds__(256)
void angle_kernel(const float* __restrict__ Gs, const float* __restrict__ Gt,
                  const float* __restrict__ invS, const float* __restrict__ invT,
                  float* __restrict__ partial) {
    __shared__ float sGs[IPB][KPB];
    __shared__ float sGt[IPB][KPB];
    __shared__ float red[256];
    const int j  = threadIdx.x;
    const int ig = blockIdx.x & 31;                  // 32 i-groups
    const int kc = blockIdx.x >> 5;                  // 8 k-chunks
    const int i0 = ig * IPB;
    const int k0 = kc * KPB;

    {   // stage block-uniform G[i][k] tiles into LDS (one load each)
        const int ii = threadIdx.x >> 5;             // 0..7
        const int kk = threadIdx.x & 31;             // 0..31
        sGs[ii][kk] = Gs[(i0 + ii) * NR + (k0 + kk)];
        sGt[ii][kk] = Gt[(i0 + ii) * NR + (k0 + kk)];
    }
    __syncthreads();

    const float gjj_s = Gs[j * NR + j];
    const float gjj_t = Gt[j * NR + j];
    float bs[IPB], bt[IPB], ws[IPB], wt[IPB];
    #pragma unroll
    for (int ii = 0; ii < IPB; ++ii) {
        const int i = i0 + ii;
        bs[ii] = gjj_s - Gs[i * NR + j];
        bt[ii] = gjj_t - Gt[i * NR + j];
        ws[ii] = invS[i * NR + j];
        wt[ii] = invT[i * NR + j];
    }

    float acc = 0.f;
    for (int kk = 0; kk < KPB; ++kk) {
        const int k = k0 + kk;
        const float gkj_s = Gs[k * NR + j];
        const float gkj_t = Gt[k * NR + j];
        const float wkj_s = invS[k * NR + j];
        const float wkj_t = invT[k * NR + j];
        const bool kvj = (k != j);
        #pragma unroll
        for (int ii = 0; ii < IPB; ++ii) {
            const int i = i0 + ii;
            float ps = (bs[ii] - gkj_s + sGs[ii][kk]) * ws[ii] * wkj_s;
            float pt = (bt[ii] - gkj_t + sGt[ii][kk]) * wt[ii] * wkj_t;
            float a = fabsf(ps - pt);
            float h = (a < 1.f) ? 0.5f * a * a : a - 0.5f;
            if (kvj && (i != j) && (k != i)) acc += h;
        }
    }

    red[j] = acc; __syncthreads();
    for (int o = 128; o > 0; o >>= 1) { if (j < o) red[j] += red[j + o]; __syncthreads(); }
    if (j == 0) partial[blockIdx.x] = red[0];
}

// -------- finalize ----------------------------------------------------------
__global__ __launch_bounds__(256)
void finalize_kernel(const float* __restrict__ partial, const float* __restrict__ scal,
                     float* __restrict__ out) {
    __shared__ float red[256];
    const int t = threadIdx.x;
    red[t] = partial[t]; __syncthreads();
    for (int o = 128; o > 0; o >>= 1) { if (t < o) red[t] += red[t + o]; __syncthreads(); }
    if (t == 0) {
        const float angle = red[0] / 16581120.f;     // 256*255*254
        const float dist  = scal[0];
        const float con   = scal[1];
        const float kd    = 0.5f * dist + 0.5f * angle;
        out[0] = con + kd;
        out[1] = con;
        out[2] = kd;
    }
}

extern "C" void kernel_launch(void* const* d_in, const int* in_sizes, int n_in,
                              void* d_out, int out_size, void* d_ws, size_t ws_size,
                              hipStream_t stream) {
    const float* sq = (const float*)d_in[0];
    const float* sp = (const float*)d_in[1];
    const float* tq = (const float*)d_in[2];
    const float* tp = (const float*)d_in[3];
    float* out = (float*)d_out;

    float* ws      = (float*)d_ws;
    float* Gs      = ws;                 // 256*256
    float* Gt      = Gs   + NR * NR;     // 256*256
    float* invS    = Gt   + NR * NR;     // 256*256
    float* invT    = invS + NR * NR;     // 256*256
    float* scores  = invT + NR * NR;     // 128*128
    float* scal    = scores + NB * NB;   // [0]=dist, [1]=contrastive
    float* partial = scal + 8;           // 256

    gram_wmma_kernel  <<<32, 256, 0, stream>>>(sq, sp, Gs);
    gram_wmma_kernel  <<<32, 256, 0, stream>>>(tq, tp, Gt);
    scores_wmma_kernel<<< 8, 256, 0, stream>>>(sq, sp, scores);
    distance_kernel   <<< 1, 1024, 0, stream>>>(Gs, Gt, invS, invT, scal);
    contrastive_kernel<<< 1, 128, 0, stream>>>(scores, scal);
    angle_kernel      <<<256, 256, 0, stream>>>(Gs, Gt, invS, invT, partial);
    finalize_kernel   <<< 1, 256, 0, stream>>>(partial, scal, out);
}